// MultiHeadAttention_77094662963762
// MI455X (gfx1250) — compile-verified
//
#include <hip/hip_runtime.h>
#include <hip/hip_bf16.h>

// ---------------------------------------------------------------------------
// MultiHeadAttention forward on gfx1250 (MI455X), fp32 via V_WMMA_F32_16X16X4_F32
// B=2, S=2048, D=1024, H=16, HD=64
// ---------------------------------------------------------------------------

typedef __attribute__((ext_vector_type(2))) float v2f;
typedef __attribute__((ext_vector_type(8))) float v8f;

#define WMMA_F32(A, Bf, C) \
    __builtin_amdgcn_wmma_f32_16x16x4_f32(false, (A), false, (Bf), (short)0, (C), false, false)

static constexpr int BB = 2;
static constexpr int SS = 2048;
static constexpr int DD = 1024;
static constexpr int HH = 16;
static constexpr int HD = 64;
static constexpr int SC_PITCH = 2052;   // 2048 + 4 pad -> conflict-free LDS column reads

// ---------------------------------------------------------------------------
// GEMM + bias: Out = X(4096xD) @ W^T(DxD) + bias.  One wave per 16x16 tile.
// headMajor=1 -> write [b][h][s][hd] (for attention); else plain row-major.
// ---------------------------------------------------------------------------
__global__ __launch_bounds__(32)
void proj_kernel(const float* __restrict__ X, const float* __restrict__ W,
                 const float* __restrict__ bias, float* __restrict__ Out,
                 int headMajor)
{
    const int lane = threadIdx.x;          // wave32
    const int mn   = lane & 15;            // A-row m / B-col n
    const int kb   = (lane >> 4) * 2;      // K-pair: lanes 0-15 -> K0,K1 ; 16-31 -> K2,K3

    const int row0 = blockIdx.y * 16;      // rows of X (b*S+s)
    const int col0 = blockIdx.x * 16;      // output feature

    const float* __restrict__ arow = X + (size_t)(row0 + mn) * DD;   // A[m][*]
    const float* __restrict__ brow = W + (size_t)(col0 + mn) * DD;   // B[k][n] = W[n][k]

    v8f acc = {};
    #pragma unroll 8
    for (int k = 0; k < DD; k += 4) {
        v2f a = { arow[k + kb], arow[k + kb + 1] };
        v2f b = { brow[k + kb], brow[k + kb + 1] };
        acc = WMMA_F32(a, b, acc);
    }

    const int   cn = col0 + mn;
    const float bv = bias[cn];
    const int   rb = row0 + ((lane >> 4) ? 8 : 0);   // C/D: M = r + 8*(lane>=16)

    #pragma unroll
    for (int r = 0; r < 8; ++r) {
        const float v  = acc[r] + bv;
        const int   rr = rb + r;                      // global row = b*S + s
        if (headMajor) {
            const int b = rr >> 11, s = rr & (SS - 1);
            const int h = cn >> 6,  hd = cn & (HD - 1);
            Out[(((size_t)(b * HH + h) * SS) + s) * HD + hd] = v;
        } else {
            Out[(size_t)rr * DD + cn] = v;
        }
    }
}

// ---------------------------------------------------------------------------
// Fused attention: one block = 16 query rows of one (b,h).
// Phase 1: scores -> LDS (4 waves split 128 key tiles, f32 WMMA)
// Phase 2: masked softmax (8 thr/row reductions), coalesced attn write to d_out
// Phase 3: ctx = W @ V (each wave owns 16 of the 64 head dims, f32 WMMA, A from LDS)
// ---------------------------------------------------------------------------
__global__ __launch_bounds__(128)
void attn_kernel(const float* __restrict__ qp, const float* __restrict__ kp,
                 const float* __restrict__ vp, const int* __restrict__ mask,
                 float* __restrict__ attnOut, float* __restrict__ ctxOut)
{
    const int b  = blockIdx.z;
    const int h  = blockIdx.y;
    const int q0 = blockIdx.x * 16;

    const float* __restrict__ Q = qp + (((size_t)(b * HH + h) * SS) + q0) * HD;
    const float* __restrict__ K = kp + (size_t)(b * HH + h) * SS * HD;
    const float* __restrict__ V = vp + (size_t)(b * HH + h) * SS * HD;

    extern __shared__ float smem[];                 // [16][SC_PITCH] + rowM[16] + rowInv[16]
    float* rowM   = smem + 16 * SC_PITCH;
    float* rowInv = rowM + 16;

    const int tid  = threadIdx.x;
    const int wave = tid >> 5;
    const int lane = tid & 31;
    const int mn   = lane & 15;
    const int kb   = (lane >> 4) * 2;
    const int rb   = (lane >> 4) * 8;

    // --- Phase 1: scores = (Q @ K^T) * scale ------------------------------
    v2f aq[16];                                     // 16x64 A fragments, reused 32x
    {
        const float* qrow = Q + mn * HD;
        #pragma unroll
        for (int kk = 0; kk < 16; ++kk)
            aq[kk] = { qrow[4 * kk + kb], qrow[4 * kk + kb + 1] };
    }
    const float scale = 0.125f;                     // 1/sqrt(64)

    for (int kt = wave; kt < SS / 16; kt += 4) {    // uniform per wave: EXEC all ones
        const float* krow = K + (size_t)(kt * 16 + mn) * HD;   // B[k][n] = K[n][k]
        v8f acc = {};
        #pragma unroll
        for (int kk = 0; kk < 16; ++kk) {
            v2f bfrag = { krow[4 * kk + kb], krow[4 * kk + kb + 1] };
            acc = WMMA_F32(aq[kk], bfrag, acc);
        }
        #pragma unroll
        for (int r = 0; r < 8; ++r)
            smem[(rb + r) * SC_PITCH + kt * 16 + mn] = acc[r] * scale;
    }
    __syncthreads();

    // --- Phase 2: masked softmax ------------------------------------------
    {
        const int row = tid >> 3;                   // 16 rows x 8 threads
        const int sub = tid & 7;
        const int* __restrict__ mrow = mask + ((size_t)b * SS + q0 + row) * SS;

        float mx = -3.0e38f;
        for (int c = sub; c < SS; c += 8) {
            float s = smem[row * SC_PITCH + c];
            if (mrow[c] == 0) s = -1.0e9f;
            smem[row * SC_PITCH + c] = s;           // store masked score
            mx = fmaxf(mx, s);
        }
        mx = fmaxf(mx, __shfl_xor(mx, 1, 8));
        mx = fmaxf(mx, __shfl_xor(mx, 2, 8));
        mx = fmaxf(mx, __shfl_xor(mx, 4, 8));

        float sum = 0.0f;
        for (int c = sub; c < SS; c += 8)
            sum += __expf(smem[row * SC_PITCH + c] - mx);
        sum += __shfl_xor(sum, 1, 8);
        sum += __shfl_xor(sum, 2, 8);
        sum += __shfl_xor(sum, 4, 8);

        if (sub == 0) { rowM[row] = mx; rowInv[row] = __frcp_rn(sum); }
    }
    __syncthreads();

    // normalize in LDS + fully coalesced write of attn weights (the big output)
    {
        float* __restrict__ abase = attnOut + (((size_t)(b * HH + h) * SS) + q0) * SS;
        for (int e = tid; e < 16 * SS; e += 128) {
            const int r = e >> 11;
            const int c = e & (SS - 1);
            const float w = __expf(smem[r * SC_PITCH + c] - rowM[r]) * rowInv[r];
            smem[r * SC_PITCH + c] = w;
            abase[(size_t)r * SS + c] = w;
        }
    }
    __syncthreads();

    // --- Phase 3: ctx = W(16x2048) @ V(2048x64); wave owns cols [16w,16w+16)
    {
        const int c0 = wave * 16;
        v8f acc = {};
        for (int k = 0; k < SS; k += 4) {
            v2f a = { smem[mn * SC_PITCH + k + kb], smem[mn * SC_PITCH + k + kb + 1] };
            v2f bfrag = { V[(size_t)(k + kb) * HD + c0 + mn],
                          V[(size_t)(k + kb + 1) * HD + c0 + mn] };
            acc = WMMA_F32(a, bfrag, acc);
        }
        // ctx stored row-major [b][s][h*HD+hd] for the output projection
        #pragma unroll
        for (int r = 0; r < 8; ++r) {
            const int s = q0 + rb + r;
            ctxOut[((size_t)(b * SS + s)) * DD + h * HD + c0 + mn] = acc[r];
        }
    }
}

// ---------------------------------------------------------------------------
extern "C" void kernel_launch(void* const* d_in, const int* in_sizes, int n_in,
                              void* d_out, int out_size, void* d_ws, size_t ws_size,
                              hipStream_t stream)
{
    const float* q    = (const float*)d_in[0];
    const float* k    = (const float*)d_in[1];
    const float* v    = (const float*)d_in[2];
    const int*   mask = (const int*)  d_in[3];
    const float* Wq   = (const float*)d_in[4];
    const float* bq   = (const float*)d_in[5];
    const float* Wout = (const float*)d_in[6];
    const float* bout = (const float*)d_in[7];

    float* out  = (float*)d_out;                        // [B,S,D]
    float* attn = out + (size_t)BB * SS * DD;           // [B,H,S,S]

    const size_t mat = (size_t)BB * SS * DD;            // 4,194,304 floats
    float* ws_q   = (float*)d_ws;                       // [B,H,S,HD]
    float* ws_k   = ws_q + mat;
    float* ws_v   = ws_k + mat;
    float* ws_ctx = ws_v + mat;                         // [B,S,D]

    dim3 gProj(DD / 16, (BB * SS) / 16);                // 64 x 256 tiles, 1 wave each
    proj_kernel<<<gProj, 32, 0, stream>>>(q, Wq, bq, ws_q, 1);
    proj_kernel<<<gProj, 32, 0, stream>>>(k, Wq, bq, ws_k, 1);
    proj_kernel<<<gProj, 32, 0, stream>>>(v, Wq, bq, ws_v, 1);

    const size_t shbytes = (size_t)(16 * SC_PITCH + 32) * sizeof(float);  // ~128.4 KB
    dim3 gAttn(SS / 16, HH, BB);
    attn_kernel<<<gAttn, 128, shbytes, stream>>>(ws_q, ws_k, ws_v, mask, attn, ws_ctx);

    proj_kernel<<<gProj, 32, 0, stream>>>(ws_ctx, Wout, bout, out, 0);
}